// WordEncoderBiRNN_89000312308180
// MI455X (gfx1250) — compile-verified
//
#include <hip/hip_runtime.h>
#include <hip/hip_bf16.h>
#include <math.h>

typedef __attribute__((ext_vector_type(2))) float v2f;
typedef __attribute__((ext_vector_type(8))) float v8f;

#define LSEQ 2048
#define HDIM 1024
#define G3H  3072   // 3*H
#define OOV  1
#define SCAN_BLOCKS 32   // 16 per direction, must all be co-resident

// ---------------------------------------------------------------------------
// Kernel 1: zero recurrent-state double buffers ([2 dirs][2 parity][1024])
// and the grid-barrier step counter (re-initialized every call -> graph-safe).
// ---------------------------------------------------------------------------
__global__ __launch_bounds__(256) void init_state_kernel(float* h, int* barrier) {
    h[blockIdx.x * 256 + threadIdx.x] = 0.0f;
    if (blockIdx.x == 0 && threadIdx.x == 0) *barrier = 0;
}

// ---------------------------------------------------------------------------
// Kernel 2: gx = X @ W_ih^T + b_ih for both directions, embedding gather +
// OOV mask fused into the A loads.  One wave computes a 32(time) x 64(row)
// strip with V_WMMA_F32_16X16X4_F32: two A fragments x four B fragments ->
// eight 16x16 fp32 accumulators (8 WMMA per 6 loads in steady state).
// ISA layouts (§7.12.2): lane = g*16 + i;  A regs hold K = 2g,2g+1 for row M=i;
// B regs hold K = 2g,2g+1 for col N=i;  C/D reg r holds M = r + 8g, N = i.
// ---------------------------------------------------------------------------
__global__ __launch_bounds__(256) void gx_gemm_wmma_kernel(
    const int* __restrict__ ids, const float* __restrict__ emb,
    const float* __restrict__ WihF, const float* __restrict__ bihF,
    const float* __restrict__ WihR, const float* __restrict__ bihR,
    float* __restrict__ gxF, float* __restrict__ gxR)
{
    const int waveGlobal = (int)((blockIdx.x * blockDim.x + threadIdx.x) >> 5);
    const int lane  = (int)(threadIdx.x & 31);
    const int i     = lane & 15;   // M-index (A) / N-index (B, C/D)
    const int g     = lane >> 4;   // K-half selector

    // Per direction: 64 time-tiles (32 rows) x 48 row-tiles (64 cols)
    const int tilesPerDir = 64 * 48;
    const int dir   = waveGlobal / tilesPerDir;       // 0 = fwd, 1 = rev
    const int tl    = waveGlobal % tilesPerDir;
    const int ttile = tl / 48;
    const int rtile = tl % 48;

    const float* __restrict__ Wih = dir ? WihR : WihF;
    const float* __restrict__ bih = dir ? bihR : bihF;
    float* __restrict__ gx        = dir ? gxR  : gxF;

    // Two A fragments: time rows [t0, t0+16) and [t0+16, t0+32)
    const int tglob0 = ttile * 32 + i;
    const int tglob1 = tglob0 + 16;
    const int srow0  = dir ? (LSEQ - 1 - tglob0) : tglob0;
    const int srow1  = dir ? (LSEQ - 1 - tglob1) : tglob1;
    const int id0 = ids[srow0];
    const int id1 = ids[srow1];
    const float amask0 = (id0 == OOV) ? 0.0f : 1.0f;
    const float amask1 = (id1 == OOV) ? 0.0f : 1.0f;
    const float* __restrict__ arow0 = emb + (size_t)id0 * HDIM + 2 * g;
    const float* __restrict__ arow1 = emb + (size_t)id1 * HDIM + 2 * g;

    const int rbase = rtile * 64;
    const float* __restrict__ brow[4];
#pragma unroll
    for (int j = 0; j < 4; ++j)
        brow[j] = Wih + (size_t)(rbase + j * 16 + i) * HDIM + 2 * g;

    v8f c0[4] = {};   // accumulators for time rows t0..t0+15
    v8f c1[4] = {};   // accumulators for time rows t0+16..t0+31

#pragma unroll 2
    for (int k0 = 0; k0 < HDIM; k0 += 4) {
        v2f a0 = amask0 * (*(const v2f*)(arow0 + k0));
        v2f a1 = amask1 * (*(const v2f*)(arow1 + k0));
        v2f b0 = *(const v2f*)(brow[0] + k0);
        v2f b1 = *(const v2f*)(brow[1] + k0);
        v2f b2 = *(const v2f*)(brow[2] + k0);
        v2f b3 = *(const v2f*)(brow[3] + k0);
        c0[0] = __builtin_amdgcn_wmma_f32_16x16x4_f32(false, a0, false, b0,
                                                      (short)0, c0[0], false, false);
        c0[1] = __builtin_amdgcn_wmma_f32_16x16x4_f32(false, a0, false, b1,
                                                      (short)0, c0[1], false, false);
        c0[2] = __builtin_amdgcn_wmma_f32_16x16x4_f32(false, a0, false, b2,
                                                      (short)0, c0[2], false, false);
        c0[3] = __builtin_amdgcn_wmma_f32_16x16x4_f32(false, a0, false, b3,
                                                      (short)0, c0[3], false, false);
        c1[0] = __builtin_amdgcn_wmma_f32_16x16x4_f32(false, a1, false, b0,
                                                      (short)0, c1[0], false, false);
        c1[1] = __builtin_amdgcn_wmma_f32_16x16x4_f32(false, a1, false, b1,
                                                      (short)0, c1[1], false, false);
        c1[2] = __builtin_amdgcn_wmma_f32_16x16x4_f32(false, a1, false, b2,
                                                      (short)0, c1[2], false, false);
        c1[3] = __builtin_amdgcn_wmma_f32_16x16x4_f32(false, a1, false, b3,
                                                      (short)0, c1[3], false, false);
    }

    // Store D + bias.  Reg r of lane (g,i): M = r + 8g (time), N = i (gate row).
#pragma unroll
    for (int j = 0; j < 4; ++j) {
        const int row  = rbase + j * 16 + i;
        const float bb = bih[row];
#pragma unroll
        for (int r = 0; r < 8; ++r) {
            const int tt0 = ttile * 32 + r + 8 * g;
            gx[(size_t)tt0 * G3H + row]        = c0[j][r] + bb;
            gx[(size_t)(tt0 + 16) * G3H + row] = c1[j][r] + bb;
        }
    }
}

// ---------------------------------------------------------------------------
// Kernel 3: persistent bidirectional GRU scan.  32 blocks x 256 threads,
// blockIdx>>4 = direction; each block owns 64 hidden units (192 gate rows).
// Per step: h_prev staged in LDS; each WAVE computes full gate rows with
// lane-strided float4 loads (coalesced 128B accesses, W_hh streams from L2),
// 5x shfl_xor cross-lane reduction; 64 threads apply the GRU gating.
// Steps are separated by a monotonic global step-counter barrier
// (release fetch_add / acquire spin + s_sleep) -- no launch overhead.
// ---------------------------------------------------------------------------
__global__ __launch_bounds__(256) void gru_scan_kernel(
    const float* __restrict__ gxF, const float* __restrict__ gxR,
    const float* __restrict__ WhhF, const float* __restrict__ bhhF,
    const float* __restrict__ WhhR, const float* __restrict__ bhhR,
    float* __restrict__ hstate,   // [2 dirs][2 parity][HDIM]
    float* __restrict__ out,      // outputs(L x 2H) ++ hidden(2H) ++ proj(2H)
    int* __restrict__ barrier)
{
    __shared__ float hs[HDIM];
    __shared__ float rowsum[192];

    const int dir   = (int)(blockIdx.x >> 4);
    const int ubase = (int)(blockIdx.x & 15) * 64;
    const int tid   = (int)threadIdx.x;
    const int wave  = tid >> 5;
    const int lane  = tid & 31;
    const int nblk  = (int)gridDim.x;

    const float* __restrict__ gx  = dir ? gxR  : gxF;
    const float* __restrict__ Whh = dir ? WhhR : WhhF;
    const float* __restrict__ bhh = dir ? bhhR : bhhF;
    float* hst = hstate + dir * 2 * HDIM;

    for (int t = 0; t < LSEQ; ++t) {
        const float* __restrict__ hprev = hst + (t & 1) * HDIM;
        float* __restrict__ hnext       = hst + ((t + 1) & 1) * HDIM;

        // Stage full h_prev of this direction into LDS (4 KB)
#pragma unroll
        for (int k = 0; k < HDIM / 256; ++k)
            hs[tid + k * 256] = hprev[tid + k * 256];
        __syncthreads();

        // 192 gate rows per block, 24 per wave; one wave computes one row
        const float4* __restrict__ h4 = (const float4*)hs;
#pragma unroll 2
        for (int j = 0; j < 24; ++j) {
            const int rloc = wave + 8 * j;           // 0..191
            const int gate = rloc >> 6;              // 0=r, 1=z, 2=n
            const int u    = ubase + (rloc & 63);    // global unit in [0,1024)
            const float4* __restrict__ w =
                (const float4*)(Whh + (size_t)(gate * HDIM + u) * HDIM);
            float s = 0.0f;
#pragma unroll
            for (int it = 0; it < 8; ++it) {
                const float4 wv = w[lane + it * 32];
                const float4 hv = h4[lane + it * 32];
                s += wv.x * hv.x + wv.y * hv.y + wv.z * hv.z + wv.w * hv.w;
            }
            s += __shfl_xor(s, 16);
            s += __shfl_xor(s, 8);
            s += __shfl_xor(s, 4);
            s += __shfl_xor(s, 2);
            s += __shfl_xor(s, 1);
            if (lane == 0) rowsum[rloc] = s;
        }
        __syncthreads();

        // GRU gating + state/output writes for this block's 64 units
        if (tid < 64) {
            const int u = ubase + tid;
            const float sr = rowsum[tid];
            const float sz = rowsum[64 + tid];
            const float sn = rowsum[128 + tid];
            const size_t gbase = (size_t)t * G3H;
            const float grx = gx[gbase + u];
            const float gzx = gx[gbase + HDIM + u];
            const float gnx = gx[gbase + 2 * HDIM + u];
            const float r = 1.0f / (1.0f + expf(-(grx + sr + bhh[u])));
            const float z = 1.0f / (1.0f + expf(-(gzx + sz + bhh[HDIM + u])));
            const float n = tanhf(gnx + r * (sn + bhh[2 * HDIM + u]));
            const float hp = hs[u];
            const float hn = (1.0f - z) * n + z * hp;
            hnext[u] = hn;
            const int orow = dir ? (LSEQ - 1 - t) : t;
            out[(size_t)orow * (2 * HDIM) + dir * HDIM + u] = hn;
            if (t == LSEQ - 1) {
                const size_t hoff = (size_t)LSEQ * (2 * HDIM);
                out[hoff + dir * HDIM + u] = hn;                 // hidden
                out[hoff + 2 * HDIM + dir * HDIM + u] = hn;      // projected
            }
        }

        // Grid barrier: make h_next visible, then wait for all 32 blocks.
        __threadfence();
        __syncthreads();
        if (tid == 0) {
            __hip_atomic_fetch_add(barrier, 1, __ATOMIC_RELEASE,
                                   __HIP_MEMORY_SCOPE_AGENT);
            const int target = (t + 1) * nblk;
            while (__hip_atomic_load(barrier, __ATOMIC_ACQUIRE,
                                     __HIP_MEMORY_SCOPE_AGENT) < target) {
                __builtin_amdgcn_s_sleep(8);
            }
        }
        __threadfence();
        __syncthreads();
    }
}

// ---------------------------------------------------------------------------
// Launcher
// ---------------------------------------------------------------------------
extern "C" void kernel_launch(void* const* d_in, const int* in_sizes, int n_in,
                              void* d_out, int out_size, void* d_ws, size_t ws_size,
                              hipStream_t stream) {
    const int*   ids  = (const int*)d_in[0];
    const float* emb  = (const float*)d_in[1];
    const float* WihF = (const float*)d_in[2];
    const float* WhhF = (const float*)d_in[3];
    const float* bihF = (const float*)d_in[4];
    const float* bhhF = (const float*)d_in[5];
    const float* WihR = (const float*)d_in[6];
    const float* WhhR = (const float*)d_in[7];
    const float* bihR = (const float*)d_in[8];
    const float* bhhR = (const float*)d_in[9];
    float* out = (float*)d_out;

    // Workspace (fp32 elems): gxF | gxR | hstate(4096) | barrier(int)
    float* gxF    = (float*)d_ws;
    float* gxR    = gxF + (size_t)LSEQ * G3H;
    float* hstate = gxR + (size_t)LSEQ * G3H;
    int*   barrier = (int*)(hstate + 4096);

    // 1) zero recurrent state + barrier counter (every call -> replay-safe)
    init_state_kernel<<<16, 256, 0, stream>>>(hstate, barrier);

    // 2) input-projection GEMMs: 2 dirs x (64 x 48) wave-tiles, 8 waves/block
    gx_gemm_wmma_kernel<<<768, 256, 0, stream>>>(ids, emb, WihF, bihF,
                                                 WihR, bihR, gxF, gxR);

    // 3) persistent serial scan, single launch, grid-barrier per timestep
    gru_scan_kernel<<<SCAN_BLOCKS, 256, 0, stream>>>(gxF, gxR,
                                                     WhhF, bhhF, WhhR, bhhR,
                                                     hstate, out, barrier);
}